// GaussianMixture_55259049230495
// MI455X (gfx1250) — compile-verified
//
#include <hip/hip_runtime.h>
#include <hip/hip_bf16.h>

// CDNA5 (gfx1250) fused Gaussian-mixture log-likelihood.
//   logits = F(N x 8) @ U(8 x M)  via V_WMMA_F32_16X16X4_F32 (two chained K=4 ops)
//   ll     = gmax + log( sum_j exp(logit - gmax) )   (gmax is a provable upper bound)

typedef __attribute__((ext_vector_type(2))) float v2f;
typedef __attribute__((ext_vector_type(8))) float v8f;

#define GMM_N 65536
#define GMM_M 1024

// ---------------------------------------------------------------------------
// Kernel 1: per-component coefficients  u_j = [c, p, r, -g00, -2g01, -g11, 0, 0]
// plus gmax = max_j (log_softmax(w)_j + 0.5*log det gamma_j).
// Single block of 1024 threads (M == 1024), LDS tree reductions.
// ---------------------------------------------------------------------------
__global__ void __launch_bounds__(GMM_M)
gmm_prep_kernel(const float* __restrict__ mu, const float* __restrict__ A,
                const float* __restrict__ w, float* __restrict__ u,
                float* __restrict__ gmax_out) {
  __shared__ float red[GMM_M];
  const int j = threadIdx.x;
  const float wj = w[j];

  // logsumexp over w (for log_softmax)
  red[j] = wj; __syncthreads();
  for (int s = GMM_M / 2; s > 0; s >>= 1) {
    if (j < s) red[j] = fmaxf(red[j], red[j + s]);
    __syncthreads();
  }
  const float wmax = red[0]; __syncthreads();
  red[j] = __expf(wj - wmax); __syncthreads();
  for (int s = GMM_M / 2; s > 0; s >>= 1) {
    if (j < s) red[j] += red[j + s];
    __syncthreads();
  }
  const float lse_w = wmax + __logf(red[0]); __syncthreads();

  // gamma_j = A_j A_j^T / 2  (2x2, symmetric)
  const float a00 = A[4 * j + 0], a01 = A[4 * j + 1];
  const float a10 = A[4 * j + 2], a11 = A[4 * j + 3];
  const float g00 = 0.5f * (a00 * a00 + a01 * a01);
  const float g01 = 0.5f * (a00 * a10 + a01 * a11);
  const float g11 = 0.5f * (a10 * a10 + a11 * a11);
  const float det = g00 * g11 - g01 * g01;
  const float logw = (wj - lse_w) + 0.5f * __logf(det);

  const float m0 = mu[2 * j + 0], m1 = mu[2 * j + 1];
  const float p   = 2.f * (g00 * m0 + g01 * m1);
  const float r   = 2.f * (g01 * m0 + g11 * m1);
  const float mqm = g00 * m0 * m0 + 2.f * g01 * m0 * m1 + g11 * m1 * m1;

  float* uj = u + 8 * j;
  uj[0] = logw - mqm;   // constant term
  uj[1] = p;            // * x0
  uj[2] = r;            // * x1
  uj[3] = -g00;         // * x0^2
  uj[4] = -2.f * g01;   // * x0*x1
  uj[5] = -g11;         // * x1^2
  uj[6] = 0.f;          // K pad
  uj[7] = 0.f;          // K pad

  // gmax: since q_ij >= 0 (gamma PSD), logit_ij <= logw_j <= gmax.
  red[j] = logw; __syncthreads();
  for (int s = GMM_M / 2; s > 0; s >>= 1) {
    if (j < s) red[j] = fmaxf(red[j], red[j + s]);
    __syncthreads();
  }
  if (j == 0) *gmax_out = red[0];
}

// ---------------------------------------------------------------------------
// Kernel 2: one wave32 per 16 sample rows. 64 column tiles of 16 components:
// two chained V_WMMA_F32_16X16X4_F32 per tile, then streaming exp-accumulate.
// A-matrix 16x4 f32 layout: lane 0-15 -> row M=lane, v0=K0, v1=K1;
//                           lane 16-31 -> row M=lane-16, v0=K2, v1=K3.
// B-matrix 4x16 f32 layout: v0 = K0 (lanes 0-15) / K2 (lanes 16-31); v1 = K1/K3.
// C/D 16x16 f32 layout: VGPR v -> rows v (lanes 0-15) and v+8 (lanes 16-31).
// ---------------------------------------------------------------------------
__global__ void __launch_bounds__(256)
gmm_lse_kernel(const float* __restrict__ sample, const float* __restrict__ u,
               const float* __restrict__ gmaxp, float* __restrict__ out) {
  const int lane = threadIdx.x & 31;
  const int wave = threadIdx.x >> 5;
  const int half = lane >> 4;          // 0: lanes 0-15, 1: lanes 16-31
  const int l15  = lane & 15;
  const int r0   = (blockIdx.x * 8 + wave) * 16;   // first sample row of tile

  // Per-row features; both halves load the same 16 rows (cache hits).
  const float2 xy = ((const float2*)sample)[r0 + l15];
  const float x0 = xy.x, x1 = xy.y;

  v2f a0, a1;                       // A fragments (K=0..3 and K=4..7)
  a0.x = half ? x1        : 1.0f;   // K2 : K0
  a0.y = half ? x0 * x0   : x0;     // K3 : K1
  a1.x = half ? 0.0f      : x0 * x1; // K6 : K4
  a1.y = half ? 0.0f      : x1 * x1; // K7 : K5

  const float gmax = *gmaxp;

  float s[8];
#pragma unroll
  for (int v = 0; v < 8; ++v) s[v] = 0.0f;

#pragma unroll 2
  for (int t = 0; t < GMM_M / 16; ++t) {
    const int j = t * 16 + l15;     // component column owned by this lane
    const float4 uA = ((const float4*)u)[2 * j + 0];  // u[0..3]
    const float4 uB = ((const float4*)u)[2 * j + 1];  // u[4..7] (6,7 are 0)

    v2f b0, b1;                     // B fragments (K=0..3 and K=4..7)
    b0.x = half ? uA.z : uA.x;      // K2 : K0
    b0.y = half ? uA.w : uA.y;      // K3 : K1
    b1.x = half ? uB.z : uB.x;      // K6 : K4
    b1.y = half ? uB.w : uB.y;      // K7 : K5

    v8f d = {};
    d = __builtin_amdgcn_wmma_f32_16x16x4_f32(false, a0, false, b0,
                                              (short)0, d, false, false);
    d = __builtin_amdgcn_wmma_f32_16x16x4_f32(false, a1, false, b1,
                                              (short)0, d, false, false);

    // streaming exp-accumulate; exp arg <= 0 by construction of gmax
#pragma unroll
    for (int v = 0; v < 8; ++v) s[v] += __expf(d[v] - gmax);
  }

  // Reduce each row's partial sums across its 16-lane half.
#pragma unroll
  for (int v = 0; v < 8; ++v) {
    float sv = s[v];
    sv += __shfl_xor(sv, 1, 32);
    sv += __shfl_xor(sv, 2, 32);
    sv += __shfl_xor(sv, 4, 32);
    sv += __shfl_xor(sv, 8, 32);
    s[v] = sv;
  }

  if (l15 == 0) {
    // VGPR v holds row r0+v (lanes 0-15) or r0+8+v (lanes 16-31).
#pragma unroll
    for (int v = 0; v < 8; ++v)
      out[r0 + half * 8 + v] = gmax + __logf(s[v]);
  }
}

// ---------------------------------------------------------------------------
extern "C" void kernel_launch(void* const* d_in, const int* in_sizes, int n_in,
                              void* d_out, int out_size, void* d_ws, size_t ws_size,
                              hipStream_t stream) {
  const float* sample = (const float*)d_in[0];   // [N, 2]
  const float* mu     = (const float*)d_in[1];   // [M, 2]
  const float* A      = (const float*)d_in[2];   // [M, 2, 2]
  const float* w      = (const float*)d_in[3];   // [M, 1]
  float* out  = (float*)d_out;                   // [N, 1]
  float* u    = (float*)d_ws;                    // [M, 8] coefficients (32 KB)
  float* gmax = u + 8 * GMM_M;                   // scalar

  gmm_prep_kernel<<<1, GMM_M, 0, stream>>>(mu, A, w, u, gmax);

  // 16 rows per wave, 8 waves per block -> 128 rows per block.
  gmm_lse_kernel<<<GMM_N / 128, 256, 0, stream>>>(sample, u, gmax, out);
}